// TrajectoryAttention_21380347199594
// MI455X (gfx1250) — compile-verified
//
#include <hip/hip_runtime.h>
#include <stdint.h>
#include <stddef.h>

// Problem constants (reference: B=8, M=32, C=1, H=W=384, DK=DV=3)
namespace {
constexpr int kB        = 8;
constexpr int kM        = 32;
constexpr int kHW       = 384 * 384;      // 147456 pixels per (b, m) slice
constexpr int kPW       = 64;             // pixels per wave tile
constexpr int kWaves    = 4;              // waves per block
constexpr int kTPB      = kWaves * 32;    // 128 threads (wave32)
constexpr int kPixBlk   = kWaves * kPW;   // 256 pixels per block
constexpr int kTilesB   = kHW / kPixBlk;  // 576 blocks per batch image (exact)
}

typedef unsigned int v4u __attribute__((ext_vector_type(4)));
typedef int          v8i __attribute__((ext_vector_type(8)));
typedef int          v4i __attribute__((ext_vector_type(4)));

#if defined(__has_builtin)
#if __has_builtin(__builtin_amdgcn_tensor_load_to_lds) && \
    __has_builtin(__builtin_amdgcn_s_wait_tensorcnt)
#define USE_TDM 1
#endif
#endif
#ifndef USE_TDM
#define USE_TDM 0
#endif

__global__ __launch_bounds__(kTPB) void traj_attn_kernel(
    const float* __restrict__ traj,  // (B, M, 1, H, W)
    const float* __restrict__ x,     // (B, 1, H, W)
    const float* __restrict__ Wq, const float* __restrict__ bq,
    const float* __restrict__ Wk, const float* __restrict__ bk,
    const float* __restrict__ Wv, const float* __restrict__ bv,
    float* __restrict__ out)         // (B, 3, H, W)
{
#if USE_TDM
  // Only shared allocation in this kernel -> assumed at LDS byte offset 0.
  __shared__ float tile[kWaves * kM * kPW];  // 4 * 32 * 64 * 4B = 32 KB
#endif
  const int tid     = threadIdx.x;
  const int wave    = tid >> 5;
  const int lane    = tid & 31;
  const int b       = blockIdx.x / kTilesB;
  const int pixWave = (blockIdx.x % kTilesB) * kPixBlk + wave * kPW;
  const int p0      = pixWave + lane * 2;    // 2 pixels per lane

#if USE_TDM
  {
    // --- Tensor DMA descriptor (ISA cdna5 §8): 2-D tile, 32 m-rows x 64 px.
    const uint64_t ga = (uint64_t)(uintptr_t)traj +
        (((uint64_t)b * kM) * (uint64_t)kHW + (uint64_t)pixWave) * 4ull;
    v4u g0;
    g0.x = 1u;                                   // count=1 (valid), user mode
    g0.y = (unsigned)(wave * kM * kPW * 4);      // lds_addr (bytes), tile at 0
    g0.z = (unsigned)(ga & 0xFFFFFFFFull);       // global_addr[31:0]
    g0.w = (unsigned)((ga >> 32) & 0x1FFFFFFull) // global_addr[56:32]
         | (2u << 30);                           // type = 2 ("image")
    v8i g1;
    g1[0] = (2 << 16);                           // data_size = 4 bytes
    g1[1] = (int)((kHW & 0xFFFF) << 16);         // tensor_dim0[15:0]
    g1[2] = (kHW >> 16) | (kM << 16);            // tensor_dim0 hi | tensor_dim1 lo
    g1[3] = (kPW << 16);                         // tile_dim0 = 64 px
    g1[4] = kM;                                  // tile_dim1 = 32, tile_dim2 = 0
    g1[5] = kHW;                                 // tensor_dim0_stride[31:0] = H*W
    g1[6] = 0;                                   // stride0 hi | stride1 lo
    g1[7] = 0;
    v4i z4 = {0, 0, 0, 0};
#if defined(__clang_major__) && __clang_major__ >= 23
    v8i z8 = {0, 0, 0, 0, 0, 0, 0, 0};
    __builtin_amdgcn_tensor_load_to_lds(g0, g1, z4, z4, z8, 0);
#else
    __builtin_amdgcn_tensor_load_to_lds(g0, g1, z4, z4, 0);
#endif
  }
#endif

  // Fold the C==1 1x1 convs + 1/sqrt(3) + log2(e) into 4 scalar coefficients:
  //   score_m * log2(e) = t_m * (A*x + C) + (B*x + D)
  const float s3l = 0.57735026918962576f * 1.44269504088896340f;
  const float wq0 = Wq[0], wq1 = Wq[1], wq2 = Wq[2];
  const float wk0 = Wk[0], wk1 = Wk[1], wk2 = Wk[2];
  const float bq0 = bq[0], bq1 = bq[1], bq2 = bq[2];
  const float bk0 = bk[0], bk1 = bk[1], bk2 = bk[2];
  const float A  = (wq0 * wk0 + wq1 * wk1 + wq2 * wk2) * s3l;
  const float Bc = (wq0 * bk0 + wq1 * bk1 + wq2 * bk2) * s3l;
  const float Cc = (bq0 * wk0 + bq1 * wk1 + bq2 * wk2) * s3l;
  const float Dc = (bq0 * bk0 + bq1 * bk1 + bq2 * bk2) * s3l;

  const float2 xv = *(const float2*)&x[(size_t)b * kHW + p0];
  const float u0 = fmaf(A, xv.x, Cc), c0 = fmaf(Bc, xv.x, Dc);
  const float u1 = fmaf(A, xv.y, Cc), c1 = fmaf(Bc, xv.y, Dc);

#if USE_TDM
  __builtin_amdgcn_s_wait_tensorcnt(0);
  // CRITICAL: escape the LDS array's address into an asm with a memory
  // clobber. The TDM engine wrote `tile` behind the compiler's back; without
  // this, GlobalOpt sees an LDS global with no stores and folds every load
  // from it to undef (which deleted both softmax loops last round). Passing
  // the pointer as an *input* makes the asm a potential writer of `tile`,
  // while reads below still use the original AS(3)-inferable pointer -> ds_load.
  {
    const float* esc = tile;
    asm volatile("" ::"v"(esc) : "memory");
  }
  const float* __restrict__ tw = &tile[wave * kM * kPW];
#endif

  // Pass 1: load all 32 trajectory samples (2 px/lane) and find row max.
  float2 tv[kM];
  float mx0 = -__builtin_inff(), mx1 = -__builtin_inff();
#pragma unroll
  for (int m = 0; m < kM; ++m) {
#if USE_TDM
    tv[m] = *(const float2*)&tw[m * kPW + lane * 2];
#else
    tv[m] = *(const float2*)&traj[((size_t)b * kM + m) * (size_t)kHW + p0];
#endif
    mx0 = fmaxf(mx0, fmaf(tv[m].x, u0, c0));
    mx1 = fmaxf(mx1, fmaf(tv[m].y, u1, c1));
  }

  // Pass 2: one exp2 per element (scores pre-scaled by log2(e) -> exact softmax).
  float l0 = 0.f, S0 = 0.f, l1 = 0.f, S1 = 0.f;
#pragma unroll
  for (int m = 0; m < kM; ++m) {
    const float e0 = exp2f(fmaf(tv[m].x, u0, c0) - mx0);
    const float e1 = exp2f(fmaf(tv[m].y, u1, c1) - mx1);
    l0 += e0;  S0 = fmaf(e0, tv[m].x, S0);
    l1 += e1;  S1 = fmaf(e1, tv[m].y, S1);
  }
  const float r0 = S0 / l0;
  const float r1 = S1 / l1;

  // out_v = wv_v * weighted_mean + bv_v   (softmax weights sum to 1)
  const float wv0 = Wv[0], wv1 = Wv[1], wv2 = Wv[2];
  const float bv0 = bv[0], bv1 = bv[1], bv2 = bv[2];
  float* o = out + (size_t)b * 3 * kHW + p0;
  *(float2*)&o[0]       = make_float2(fmaf(wv0, r0, bv0), fmaf(wv0, r1, bv0));
  *(float2*)&o[kHW]     = make_float2(fmaf(wv1, r0, bv1), fmaf(wv1, r1, bv1));
  *(float2*)&o[2 * kHW] = make_float2(fmaf(wv2, r0, bv2), fmaf(wv2, r1, bv2));
}

extern "C" void kernel_launch(void* const* d_in, const int* in_sizes, int n_in,
                              void* d_out, int out_size, void* d_ws, size_t ws_size,
                              hipStream_t stream) {
  (void)in_sizes; (void)n_in; (void)out_size; (void)d_ws; (void)ws_size;
  const float* traj = (const float*)d_in[0];
  const float* x    = (const float*)d_in[1];
  const float* Wq   = (const float*)d_in[2];
  const float* bq   = (const float*)d_in[3];
  const float* Wk   = (const float*)d_in[4];
  const float* bk   = (const float*)d_in[5];
  const float* Wv   = (const float*)d_in[6];
  const float* bv   = (const float*)d_in[7];
  float* out = (float*)d_out;

  const int grid = kB * kTilesB;  // 4608 blocks, each 4 waves x 64 px
  traj_attn_kernel<<<dim3(grid), dim3(kTPB), 0, stream>>>(
      traj, x, Wq, bq, Wk, bk, Wv, bv, out);
}